// GNNModel_25297357373522
// MI455X (gfx1250) — compile-verified
//
#include <hip/hip_runtime.h>
#include <hip/hip_bf16.h>

typedef __attribute__((ext_vector_type(2))) float v2f;
typedef __attribute__((ext_vector_type(8))) float v8f;

#define NNODES 50000
#define NEDGES 800000
#define DH     128
#define DOUT   64

// ---------------------------------------------------------------------------
// Degree / normalization
// ---------------------------------------------------------------------------
__global__ void deg_init_kernel(float* __restrict__ deg, int n) {
    int t = blockIdx.x * blockDim.x + threadIdx.x;
    if (t < n) deg[t] = 1.0f;                      // self-loop contributes 1
}

__global__ void deg_accum_kernel(const int* __restrict__ dst,
                                 float* __restrict__ deg, int e) {
    int t = blockIdx.x * blockDim.x + threadIdx.x;
    if (t < e) atomicAdd(&deg[dst[t]], 1.0f);
}

__global__ void deg_rsqrt_kernel(float* __restrict__ deg, int n) {
    int t = blockIdx.x * blockDim.x + threadIdx.x;
    if (t < n) deg[t] = rsqrtf(deg[t]);            // deg >= 1 always
}

// ---------------------------------------------------------------------------
// GEMM: out[N,128] = A[N,128] @ B[128,128], fp32 WMMA 16x16x4.
// One wave per 16x16 tile; 8 waves/block cover 16 rows x all 128 cols.
// A tile staged in LDS (padded stride 132 to avoid 64-bank conflicts).
// Requires nrows % 16 == 0 (50000 = 3125*16).
// ---------------------------------------------------------------------------
__global__ __launch_bounds__(256) void gemm_wmma_f32_kernel(
        const float* __restrict__ A, const float* __restrict__ B,
        float* __restrict__ out) {
    __shared__ float sA[16 * 132];

    const int t    = threadIdx.x;
    const int row0 = blockIdx.x * 16;

    // Cooperative load of 16x128 A tile (coalesced float4 per thread).
    {
        const int c4 = (t & 31) * 4;
        const int r0 = t >> 5;                     // 0..7
        #pragma unroll
        for (int rr = r0; rr < 16; rr += 8) {
            float4 v = *(const float4*)(A + (size_t)(row0 + rr) * DH + c4);
            *(float4*)(&sA[rr * 132 + c4]) = v;
        }
    }
    __syncthreads();

    const int wave    = t >> 5;                    // 0..7 -> tile column
    const int lane    = t & 31;
    const int l15     = lane & 15;
    const int halfsel = lane >> 4;                 // K-half select
    const int col0    = wave * 16;
    const int bcol    = col0 + l15;

    const float* __restrict__ bp = B + bcol;       // column pointer into W
    v8f acc = {};

    #pragma unroll
    for (int k = 0; k < DH; k += 4) {
        const int kk = k + halfsel * 2;
        v2f a = *(const v2f*)(&sA[l15 * 132 + kk]);          // A[m=l15][kk,kk+1]
        v2f b;
        b.x = bp[(size_t)kk * DH];                           // B[kk][bcol]
        b.y = bp[(size_t)(kk + 1) * DH];                     // B[kk+1][bcol]
        acc = __builtin_amdgcn_wmma_f32_16x16x4_f32(
                  false, a, false, b, (short)0, acc, false, false);
    }

    // D layout: vgpr r holds row M = r + 8*halfsel, col N = l15.
    float* __restrict__ orow = out + (size_t)row0 * DH + col0;
    #pragma unroll
    for (int r = 0; r < 8; ++r) {
        const int m = r + 8 * halfsel;
        orow[(size_t)m * DH + l15] = acc[r];
    }
}

// ---------------------------------------------------------------------------
// Aggregation: agg = self-loop term, then atomic scatter of edge messages.
// ---------------------------------------------------------------------------
__global__ void agg_selfloop_kernel(const float* __restrict__ hw,
                                    const float* __restrict__ dinv,
                                    float* __restrict__ agg, int total) {
    int t = blockIdx.x * blockDim.x + threadIdx.x;
    if (t < total) {
        int i = t >> 7;                            // node
        float w = dinv[i] * dinv[i];               // self-loop norm
        agg[t] = w * hw[t];
    }
}

// One wave32 per edge; each lane moves 4 features (float4 gather + 4 atomics).
__global__ __launch_bounds__(256) void edge_scatter_kernel(
        const int* __restrict__ src, const int* __restrict__ dst,
        const float* __restrict__ dinv, const float* __restrict__ hw,
        float* __restrict__ agg, int e) {
    int t    = blockIdx.x * blockDim.x + threadIdx.x;
    int edge = t >> 5;
    int lane = t & 31;
    if (edge >= e) return;
    int s = src[edge];
    int d = dst[edge];
    float w = dinv[s] * dinv[d];
    float4 v = *(const float4*)(hw + (size_t)s * DH + lane * 4);
    float* a = agg + (size_t)d * DH + lane * 4;
    atomicAdd(a + 0, w * v.x);
    atomicAdd(a + 1, w * v.y);
    atomicAdd(a + 2, w * v.z);
    atomicAdd(a + 3, w * v.w);
}

__global__ void bias_relu_kernel(float* __restrict__ h,
                                 const float* __restrict__ bias, int total) {
    int t = blockIdx.x * blockDim.x + threadIdx.x;
    if (t < total) {
        float v = h[t] + bias[t & (DH - 1)];
        h[t] = v > 0.0f ? v : 0.0f;
    }
}

// ---------------------------------------------------------------------------
// Global max pool over nodes. h >= 0 (post-ReLU), so uint bit compare == float
// compare; per-block register reduction then one atomicMax per (block,feat).
// ---------------------------------------------------------------------------
__global__ void pool_zero_kernel(unsigned* __restrict__ pooled) {
    pooled[threadIdx.x] = 0u;                      // == __float_as_uint(0.0f)
}

__global__ __launch_bounds__(128) void maxpool_kernel(
        const float* __restrict__ h, unsigned* __restrict__ pooled, int n) {
    int f     = threadIdx.x;                       // feature 0..127
    int start = blockIdx.x * 256;
    int end   = start + 256; if (end > n) end = n;
    float m = 0.0f;
    for (int i = start; i < end; ++i)
        m = fmaxf(m, h[(size_t)i * DH + f]);
    atomicMax(&pooled[f], __float_as_uint(m));
}

__global__ void fc_out_kernel(const unsigned* __restrict__ pooled,
                              const float* __restrict__ Wfc,
                              const float* __restrict__ bfc,
                              float* __restrict__ out) {
    int j = threadIdx.x;                           // 0..63
    float s = bfc[j];
    #pragma unroll 4
    for (int f = 0; f < DH; ++f)
        s += __uint_as_float(pooled[f]) * Wfc[f * DOUT + j];
    out[j] = s;
}

// ---------------------------------------------------------------------------
extern "C" void kernel_launch(void* const* d_in, const int* in_sizes, int n_in,
                              void* d_out, int out_size, void* d_ws, size_t ws_size,
                              hipStream_t stream) {
    (void)in_sizes; (void)n_in; (void)out_size; (void)ws_size;

    const float* x   = (const float*)d_in[0];
    const int*   ei  = (const int*)  d_in[1];
    const float* W1  = (const float*)d_in[2];
    const float* b1  = (const float*)d_in[3];
    const float* W2  = (const float*)d_in[4];
    const float* b2  = (const float*)d_in[5];
    const float* Wfc = (const float*)d_in[6];
    const float* bfc = (const float*)d_in[7];
    float*       out = (float*)d_out;

    const int* src = ei;               // edge_index[0]
    const int* dst = ei + NEDGES;      // edge_index[1]

    // Workspace layout (floats): dinv[N] | bufA[N*128] | bufB[N*128] | pooled[128]
    float*    ws     = (float*)d_ws;
    float*    dinv   = ws;
    float*    bufA   = dinv + NNODES;                         // hw (GEMM out)
    float*    bufB   = bufA + (size_t)NNODES * DH;            // agg / h
    unsigned* pooled = (unsigned*)(bufB + (size_t)NNODES * DH);

    const int total   = NNODES * DH;
    const int gNodes  = (NNODES + 255) / 256;
    const int gEdges  = (NEDGES + 255) / 256;
    const int gTotal  = (total + 255) / 256;
    const int gEdgeW  = (NEDGES * 32 + 255) / 256;            // wave per edge
    const int gTiles  = NNODES / 16;                          // 3125

    // --- normalization ---
    deg_init_kernel  <<<gNodes, 256, 0, stream>>>(dinv, NNODES);
    deg_accum_kernel <<<gEdges, 256, 0, stream>>>(dst, dinv, NEDGES);
    deg_rsqrt_kernel <<<gNodes, 256, 0, stream>>>(dinv, NNODES);

    // --- layer 1: hw = x@W1 ; agg ; relu(agg+b1) -> bufB ---
    gemm_wmma_f32_kernel <<<gTiles, 256, 0, stream>>>(x, W1, bufA);
    agg_selfloop_kernel  <<<gTotal, 256, 0, stream>>>(bufA, dinv, bufB, total);
    edge_scatter_kernel  <<<gEdgeW, 256, 0, stream>>>(src, dst, dinv, bufA, bufB, NEDGES);
    bias_relu_kernel     <<<gTotal, 256, 0, stream>>>(bufB, b1, total);

    // --- layer 2: hw = h1@W2 ; agg ; relu(agg+b2) -> bufB ---
    gemm_wmma_f32_kernel <<<gTiles, 256, 0, stream>>>(bufB, W2, bufA);
    agg_selfloop_kernel  <<<gTotal, 256, 0, stream>>>(bufA, dinv, bufB, total);
    edge_scatter_kernel  <<<gEdgeW, 256, 0, stream>>>(src, dst, dinv, bufA, bufB, NEDGES);
    bias_relu_kernel     <<<gTotal, 256, 0, stream>>>(bufB, b2, total);

    // --- max pool + FC head ---
    pool_zero_kernel <<<1, DH, 0, stream>>>(pooled);
    maxpool_kernel   <<<(NNODES + 255) / 256, DH, 0, stream>>>(bufB, pooled, NNODES);
    fc_out_kernel    <<<1, DOUT, 0, stream>>>(pooled, Wfc, bfc, out);
}